// CapsuleLayer1d_41601053229673
// MI455X (gfx1250) — compile-verified
//
#include <hip/hip_runtime.h>

// CDNA5 / gfx1250 fused capsule routing kernel, round 2.
// B=4096, O=32, N=64, DOUT=DIN=16, 3 routing iterations.
// One workgroup = 16 batch elements, 1024 threads = 32 wave32, one out-capsule per wave.
// u_hat never touches global memory; v kept in registers; reductions via DPP.

typedef __attribute__((ext_vector_type(2))) float v2f;
typedef __attribute__((ext_vector_type(4))) float f4;
typedef __attribute__((ext_vector_type(8))) float v8f;

#define O_CAPS 32
#define N_CAPS 64
#define DOUT 16
#define DIN 16
#define BT 16            // batch tile per workgroup
#define THREADS 1024     // 32 wave32

// ---- LDS layout (in floats) ----
#define XS_OFF   0                        // x tile: [n=64][i=16][b padded to 17]
#define XS_SZ    (64 * 16 * 17)
#define BL_OFF   (XS_OFF + XS_SZ)         // routing logits: [o=32][n=64][b=16]
#define BL_SZ    (32 * 64 * 16)
#define MX_OFF   (BL_OFF + BL_SZ)         // softmax max:   [n=64][b=16]
#define MX_SZ    (64 * 16)
#define RS_OFF   (MX_OFF + MX_SZ)         // softmax 1/sum: [n=64][b=16]
#define RS_SZ    (64 * 16)
#define LDS_FLOATS (RS_OFF + RS_SZ)       // 52224 floats = 208896 bytes (< 320 KB)

// DPP8 selector: lane i reads lane sel[i] within each 8-lane group
#define DPP8(s0,s1,s2,s3,s4,s5,s6,s7) \
    ((s0)|((s1)<<3)|((s2)<<6)|((s3)<<9)|((s4)<<12)|((s5)<<15)|((s6)<<18)|((s7)<<21))

// Butterfly sum over the 16-lane half-wave group (lanes 0-15 / 16-31 separately):
// xor1, xor2, xor4 via DPP8; xor8 via DPP16 row_ror:8 (rotate by 8 within 16-lane row).
__device__ __forceinline__ float red16_sum(float x) {
    x += __int_as_float(__builtin_amdgcn_mov_dpp8(__float_as_int(x),
            DPP8(1,0,3,2,5,4,7,6)));                              // xor 1
    x += __int_as_float(__builtin_amdgcn_mov_dpp8(__float_as_int(x),
            DPP8(2,3,0,1,6,7,4,5)));                              // xor 2
    x += __int_as_float(__builtin_amdgcn_mov_dpp8(__float_as_int(x),
            DPP8(4,5,6,7,0,1,2,3)));                              // xor 4
    x += __int_as_float(__builtin_amdgcn_mov_dpp(__float_as_int(x),
            0x128, 0xf, 0xf, true));                              // row_ror:8 == xor 8
    return x;
}

extern "C" __global__ __launch_bounds__(THREADS)
void caps_routing_kernel(const float* __restrict__ xg,
                         const float* __restrict__ wg,
                         float* __restrict__ outg)
{
    extern __shared__ float lds[];
    float* xs   = lds + XS_OFF;
    float* blog = lds + BL_OFF;
    float* mx   = lds + MX_OFF;
    float* rs   = lds + RS_OFF;

    const int t    = threadIdx.x;
    const int wave = t >> 5;      // = out-capsule o
    const int lane = t & 31;
    const int bl   = lane & 15;   // A-row (batch) / B-col (d) index for WMMA lanes
    const int half = lane >> 4;   // selects K pair within WMMA fragments
    const int bb   = blockIdx.x;  // batch tile index (0..255)

    // ---- stage x tile [16][64][16] -> LDS xs[n][i][b17]; zero the logits ----
    {
        const f4* xv = (const f4*)(xg + (size_t)bb * BT * N_CAPS * DIN);
        #pragma unroll
        for (int k = 0; k < 4; ++k) {
            int idx = k * THREADS + t;          // float4 slot: b*256 + n*4 + i4
            int b   = idx >> 8;
            int rem = idx & 255;
            int n   = rem >> 2;
            int i4  = rem & 3;
            f4 v = xv[idx];
            float* dst = xs + (n * 16 + i4 * 4) * 17 + b;
            dst[0 * 17] = v.x; dst[1 * 17] = v.y; dst[2 * 17] = v.z; dst[3 * 17] = v.w;
        }
        #pragma unroll
        for (int k = 0; k < BL_SZ / THREADS; ++k)
            blog[k * THREADS + t] = 0.0f;
    }
    __syncthreads();

    const int o = wave;
    const float* wo = wg + (size_t)o * N_CAPS * DOUT * DIN;

    for (int it = 0; it < 3; ++it) {
        // ---- softmax stats over o for every (n, b): max and 1/sum(exp) ----
        {
            int n = t >> 4, b = t & 15;         // 1024 (n,b) pairs, one per thread
            float m = -1e30f;
            for (int oo = 0; oo < O_CAPS; ++oo)
                m = fmaxf(m, blog[(oo * N_CAPS + n) * 16 + b]);
            float ssum = 0.0f;
            for (int oo = 0; oo < O_CAPS; ++oo)
                ssum += __expf(blog[(oo * N_CAPS + n) * 16 + b] - m);
            mx[n * 16 + b] = m;
            rs[n * 16 + b] = 1.0f / ssum;
        }
        __syncthreads();

        const bool last = (it == 2);

        // ---- s[b,d] = sum_{n,i} (x[b,n,i]*c[b,o,n]) * W[o,n,d,i] via WMMA ----
        v8f acc = {};
        for (int n = 0; n < N_CAPS; ++n) {
            float c = __expf(blog[(o * N_CAPS + n) * 16 + bl] - mx[n * 16 + bl])
                      * rs[n * 16 + bl];
            const float* xrow = xs + n * 16 * 17 + bl;          // xs[n][i][bl]
            const float* wrow = wo + n * DOUT * DIN + bl * DIN; // W[o][n][d=bl][i]
            #pragma unroll
            for (int kc = 0; kc < 4; ++kc) {
                int i0 = kc * 4 + 2 * half;
                v2f a; a.x = xrow[i0 * 17] * c; a.y = xrow[(i0 + 1) * 17] * c;
                v2f bm = *(const v2f*)(wrow + i0);
                acc = __builtin_amdgcn_wmma_f32_16x16x4_f32(
                    false, a, false, bm, (short)0, acc, false, false);
            }
        }

        // ---- squash: acc[r] holds s[b=r+8*half][d=bl] ----
        float vreg[8];
        #pragma unroll
        for (int r = 0; r < 8; ++r) {
            float sv = acc[r];
            float nr = red16_sum(sv * sv);
            float scale = nr / ((1.0f + nr) * (__builtin_sqrtf(nr) + 1e-8f));
            vreg[r] = sv * scale;
        }

        if (last) {
            // final outputs: out[bglobal][o][d]
            float* op = outg + (((size_t)bb * BT) * O_CAPS + o) * DOUT + bl;
            #pragma unroll
            for (int r = 0; r < 8; ++r)
                op[(size_t)(r + 8 * half) * O_CAPS * DOUT] = vreg[r];
        } else {
            // ---- agreement: blog[o][n][b] += sum_d uh[b,o,n,d] * v[b,o,d] ----
            // v stays in vreg[] — same lane mapping as the uh accumulator tile.
            for (int n = 0; n < N_CAPS; ++n) {
                v8f uh = {};
                const float* xrow = xs + n * 16 * 17 + bl;
                const float* wrow = wo + n * DOUT * DIN + bl * DIN;
                #pragma unroll
                for (int kc = 0; kc < 4; ++kc) {
                    int i0 = kc * 4 + 2 * half;
                    v2f a; a.x = xrow[i0 * 17]; a.y = xrow[(i0 + 1) * 17];
                    v2f bm = *(const v2f*)(wrow + i0);
                    uh = __builtin_amdgcn_wmma_f32_16x16x4_f32(
                        false, a, false, bm, (short)0, uh, false, false);
                }
                #pragma unroll
                for (int r = 0; r < 8; ++r) {
                    float p = red16_sum(uh[r] * vreg[r]);
                    if (bl == r)   // lanes r / 16+r commit rows b=r / b=r+8
                        blog[(o * N_CAPS + n) * 16 + (r + 8 * half)] += p;
                }
            }
        }
        __syncthreads();
    }
}

extern "C" void kernel_launch(void* const* d_in, const int* in_sizes, int n_in,
                              void* d_out, int out_size, void* d_ws, size_t ws_size,
                              hipStream_t stream) {
    (void)in_sizes; (void)n_in; (void)d_ws; (void)ws_size; (void)out_size;
    const float* x = (const float*)d_in[0];   // [4096, 64, 16] fp32
    const float* w = (const float*)d_in[1];   // [32, 64, 16, 16] fp32
    float* out = (float*)d_out;               // [4096, 32, 16] fp32

    dim3 grid(4096 / BT);     // 256 workgroups
    dim3 block(THREADS);      // 32 wave32
    size_t shmem = (size_t)LDS_FLOATS * sizeof(float);  // 208896 B
    caps_routing_kernel<<<grid, block, shmem, stream>>>(x, w, out);
}